// LSTMFinder_90417651516529
// MI455X (gfx1250) — compile-verified
//
#include <hip/hip_runtime.h>
#include <hip/hip_bf16.h>
#include <math.h>

// ---------------------------------------------------------------------------
// LSTMFinder beam search on MI455X (gfx1250, wave32).
// fp32 end-to-end; GEMMs via V_WMMA_F32_16X16X4_F32 fed from LDS tiles.
// Problem sizes (fixed by reference): E=256, T=8, B=32, W=32, C=64.
// M = B*W = 1024 "beam rows" throughout.
// ---------------------------------------------------------------------------

#define E   256
#define T_  8
#define Bq  32
#define Wb  32
#define Cc  64
#define Mrows (Bq * Wb)          // 1024

typedef __attribute__((ext_vector_type(2))) float v2f;
typedef __attribute__((ext_vector_type(8))) float v8f;

// ---------------------------------------------------------------------------
// LDS-tiled fp32 WMMA GEMM:  D = act(A @ B + bias)
//   A: MxK row-major, B: KxN row-major, D: MxN row-major.
//   Workgroup = 256 threads (8 waves) -> one 64x64 output tile.
//   K swept in 64-wide panels staged in LDS via float4 (b128) fills.
//   Each wave computes two 16x16 sub-tiles (same N-sub) sharing one B
//   fragment; the two WMMAs are independent -> co-execute, no hazard NOPs.
//   Fragment layouts per CDNA5 ISA 7.12.2:
//     A 16x4 : lanes 0-15 rows M, VGPR0/1 = K0/K1 ; lanes 16-31 = K2/K3
//     B 4x16 : VGPR0 = row K0 (lanes 0-15) / K2 (lanes 16-31), VGPR1 = K1/K3
//     C/D    : VGPR r, lanes 0-15 -> M=r, lanes 16-31 -> M=r+8
// ---------------------------------------------------------------------------
#define TM  64
#define TN  64
#define KC  64
#define LDP 68           // padded row: 68 floats = 272 B (16B-aligned rows,
                         // bank stride 4 -> conflict-spread lane patterns)

__global__ __launch_bounds__(256)
void gemm_bias_act_kernel(const float* __restrict__ A,
                          const float* __restrict__ B,
                          const float* __restrict__ bias,
                          float* __restrict__ D,
                          int M, int N, int K, int relu)
{
    __shared__ float As[TM * LDP];   // m-major: As[m*LDP + k]
    __shared__ float Bs[KC * LDP];   // k-major: Bs[k*LDP + n]

    const int tid    = threadIdx.x;
    const int lane   = tid & 31;
    const int wave   = tid >> 5;            // 0..7
    const int tilesN = N / TN;
    const int bm     = (blockIdx.x / tilesN) * TM;
    const int bn     = (blockIdx.x % tilesN) * TN;

    const int half = lane >> 4;             // 0/1
    const int l16  = lane & 15;
    const int koff = half * 2;

    // wave -> sub-tile coords in the 4x4 grid of 16x16 tiles
    const int tnw = (wave & 3) * 16;        // 0,16,32,48
    const int tmw = (wave >> 2) * 16;       // 0,16 ; second tile at +32

    v8f acc0 = {0.f,0.f,0.f,0.f,0.f,0.f,0.f,0.f};
    v8f acc1 = {0.f,0.f,0.f,0.f,0.f,0.f,0.f,0.f};

    // fill-time coords: 1024 float4 per panel, 4 per thread
    for (int kk = 0; kk < K; kk += KC) {
#pragma unroll
        for (int it = 0; it < 4; ++it) {
            const int idx = it * 256 + tid;     // 0..1023
            const int r   = idx >> 4;           // 0..63
            const int q   = (idx & 15) << 2;    // 0,4,...,60 (floats)
            const float4 av = *(const float4*)(A + (size_t)(bm + r) * K + kk + q);
            *(float4*)(&As[r * LDP + q]) = av;
            const float4 bv = *(const float4*)(B + (size_t)(kk + r) * N + bn + q);
            *(float4*)(&Bs[r * LDP + q]) = bv;
        }
        __syncthreads();

#pragma unroll 4
        for (int k = 0; k < KC; k += 4) {
            v2f b, a0, a1;
            b.x  = Bs[(k + koff)     * LDP + tnw + l16];
            b.y  = Bs[(k + koff + 1) * LDP + tnw + l16];
            a0.x = As[(tmw + l16) * LDP + k + koff];
            a0.y = As[(tmw + l16) * LDP + k + koff + 1];
            a1.x = As[(tmw + 32 + l16) * LDP + k + koff];
            a1.y = As[(tmw + 32 + l16) * LDP + k + koff + 1];
            acc0 = __builtin_amdgcn_wmma_f32_16x16x4_f32(
                       false, a0, false, b, (short)0, acc0, false, false);
            acc1 = __builtin_amdgcn_wmma_f32_16x16x4_f32(
                       false, a1, false, b, (short)0, acc1, false, false);
        }
        __syncthreads();
    }

    const float bv = bias ? bias[bn + tnw + l16] : 0.f;
#pragma unroll
    for (int r = 0; r < 8; ++r) {
        const int row0 = bm + tmw + r + half * 8;
        float v0 = acc0[r] + bv;
        if (relu) v0 = fmaxf(v0, 0.f);
        D[(size_t)row0 * N + bn + tnw + l16] = v0;
        const int row1 = row0 + 32;
        float v1 = acc1[r] + bv;
        if (relu) v1 = fmaxf(v1, 0.f);
        D[(size_t)row1 * N + bn + tnw + l16] = v1;
    }
}

// ---------------------------------------------------------------------------
// Pack [Wi (512x1024) ; Wh (256x1024)] -> Wcat (768x1024), once per launch.
// ---------------------------------------------------------------------------
__global__ __launch_bounds__(256)
void pack_w_kernel(const float* __restrict__ Wi, const float* __restrict__ Wh,
                   float* __restrict__ Wcat)
{
    int i = blockIdx.x * blockDim.x + threadIdx.x;    // over 768*1024
    if (i >= 768 * 1024) return;
    int k = i >> 10, n = i & 1023;
    Wcat[i] = (k < 512) ? Wi[(size_t)k * 1024 + n]
                        : Wh[(size_t)(k - 512) * 1024 + n];
}

// ---------------------------------------------------------------------------
// Init: h0 = c0 = 0, cur0[b,w] = start_ent[b].
// ---------------------------------------------------------------------------
__global__ __launch_bounds__(256)
void init_kernel(const int* __restrict__ start_ent,
                 float* __restrict__ h, float* __restrict__ c,
                 int* __restrict__ cur)
{
    int i = blockIdx.x * blockDim.x + threadIdx.x;
    if (i < Mrows * E) { h[i] = 0.f; c[i] = 0.f; }
    if (i < Mrows)     cur[i] = start_ent[i >> 5];
}

// ---------------------------------------------------------------------------
// feat_in[m] = concat(ent_emb[cur[m]], h[m])  -> fin [1024 x 512], b128 copies
// ---------------------------------------------------------------------------
__global__ __launch_bounds__(64)
void build_fin_kernel(const float* __restrict__ ent_emb,
                      const int* __restrict__ cur,
                      const float* __restrict__ h,
                      float* __restrict__ fin)
{
    const int m = blockIdx.x;
    const int t = threadIdx.x;                       // 0..63 (float4 lanes)
    const float4* __restrict__ ep4 = (const float4*)(ent_emb + (size_t)cur[m] * E);
    const float4* __restrict__ h4  = (const float4*)(h + (size_t)m * E);
    float4* __restrict__ f4 = (float4*)(fin + (size_t)m * 512);
    f4[t]      = ep4[t];
    f4[64 + t] = h4[t];
}

// ---------------------------------------------------------------------------
// scores + softmax, one workgroup (8 waves) per beam row m.
//   score[c] = dot(rel_emb[cr[m,c]], feat[m,0:256]) +
//              dot(ent_emb[ce[m,c]], feat[m,256:512])
// Wave-per-candidate streaming dot; lanes read the 1KB rows as 2x float4
// (coalesced b128 -> this is the HBM-roofline phase). shfl_xor reduction,
// softmax by wave 0.
// ---------------------------------------------------------------------------
__device__ __forceinline__ float dot4(float4 a, float4 b) {
    return fmaf(a.x, b.x, fmaf(a.y, b.y, fmaf(a.z, b.z, a.w * b.w)));
}

__global__ __launch_bounds__(256)
void scores_softmax_kernel(const float* __restrict__ ent_emb,
                           const float* __restrict__ rel_emb,
                           const int* __restrict__ cand_rel_t,
                           const int* __restrict__ cand_ent_t,
                           const float* __restrict__ feat,
                           float* __restrict__ probs)
{
    const int m    = blockIdx.x;          // beam row 0..1023
    const int wave = threadIdx.x >> 5;    // 0..7
    const int lane = threadIdx.x & 31;

    __shared__ float4 s_feat[128];        // 512 floats
    __shared__ float  s_sc[Cc];

    {
        const float4* __restrict__ f4 = (const float4*)(feat + (size_t)m * 512);
        if (threadIdx.x < 128) s_feat[threadIdx.x] = f4[threadIdx.x];
    }
    __syncthreads();

    // this lane's slices of feat (registers, reused across candidates)
    const float4 fr0 = s_feat[lane];           // feat[  0:256) part 1
    const float4 fr1 = s_feat[32 + lane];      //               part 2
    const float4 fe0 = s_feat[64 + lane];      // feat[256:512) part 1
    const float4 fe1 = s_feat[96 + lane];      //               part 2

    for (int c = wave; c < Cc; c += 8) {
        const int ri = cand_rel_t[m * Cc + c];
        const int ei = cand_ent_t[m * Cc + c];
        const float4* __restrict__ rp4 = (const float4*)(rel_emb + (size_t)ri * E);
        const float4* __restrict__ ep4 = (const float4*)(ent_emb + (size_t)ei * E);

        // prefetch next candidate's entity row while we chew on this one
        const int cn = c + 8;
        if (cn < Cc) {
            __builtin_prefetch(ent_emb + (size_t)cand_ent_t[m * Cc + cn] * E
                               + lane * 8, 0, 0);
        }

        const float4 r0 = rp4[lane];
        const float4 r1 = rp4[32 + lane];
        const float4 e0 = ep4[lane];
        const float4 e1 = ep4[32 + lane];

        float sum = dot4(r0, fr0) + dot4(r1, fr1) + dot4(e0, fe0) + dot4(e1, fe1);
        for (int off = 16; off > 0; off >>= 1)
            sum += __shfl_xor(sum, off, 32);
        if (lane == 0) s_sc[c] = sum;
    }
    __syncthreads();

    if (wave == 0) {
        float v0 = s_sc[lane], v1 = s_sc[lane + 32];
        float mx = fmaxf(v0, v1);
        for (int off = 16; off > 0; off >>= 1)
            mx = fmaxf(mx, __shfl_xor(mx, off, 32));
        float e0 = __expf(v0 - mx), e1 = __expf(v1 - mx);
        float s = e0 + e1;
        for (int off = 16; off > 0; off >>= 1)
            s += __shfl_xor(s, off, 32);
        const float inv = 1.f / s;
        probs[(size_t)m * Cc + lane]      = e0 * inv;
        probs[(size_t)m * Cc + lane + 32] = e1 * inv;
    }
}

// ---------------------------------------------------------------------------
// Per-query top-32 of W*C = 2048 probs (iterative argmax in LDS, ties -> lower
// index, matching jax.lax.top_k ordering). Also resolves cur_next.
// ---------------------------------------------------------------------------
__global__ __launch_bounds__(256)
void topk_kernel(const float* __restrict__ probs,
                 const int* __restrict__ cand_ent_t,
                 int* __restrict__ parent, int* __restrict__ choice,
                 int* __restrict__ cur_next)
{
    const int b = blockIdx.x;                        // query 0..31
    __shared__ float s_p[Wb * Cc];                   // 2048
    __shared__ float s_rv[256];
    __shared__ int   s_ri[256];

    for (int i = threadIdx.x; i < Wb * Cc; i += 256)
        s_p[i] = probs[(size_t)b * Wb * Cc + i];
    __syncthreads();

    for (int sel = 0; sel < Wb; ++sel) {
        float bv = -1.f; int bi = 0;
        for (int i = threadIdx.x; i < Wb * Cc; i += 256) {
            float v = s_p[i];
            if (v > bv) { bv = v; bi = i; }
        }
        s_rv[threadIdx.x] = bv; s_ri[threadIdx.x] = bi;
        __syncthreads();
        for (int off = 128; off > 0; off >>= 1) {
            if (threadIdx.x < off) {
                float ov = s_rv[threadIdx.x + off];
                int   oi = s_ri[threadIdx.x + off];
                if (ov > s_rv[threadIdx.x] ||
                    (ov == s_rv[threadIdx.x] && oi < s_ri[threadIdx.x])) {
                    s_rv[threadIdx.x] = ov; s_ri[threadIdx.x] = oi;
                }
            }
            __syncthreads();
        }
        if (threadIdx.x == 0) {
            const int idx = s_ri[0];
            const int p   = idx >> 6;                // / C
            const int ch  = idx & (Cc - 1);          // % C
            const int mrow = b * Wb + sel;
            parent[mrow] = p;
            choice[mrow] = ch;
            cur_next[mrow] = cand_ent_t[(b * Wb + p) * Cc + ch];
            s_p[idx] = -1e30f;
        }
        __syncthreads();
    }
}

// ---------------------------------------------------------------------------
// xcat[m] = concat(rel_emb[cr_sel], ent_emb[ce_sel], h_prev[parent beam]),
// and c_par[m] = c_prev[parent beam]. b128 gather copies.
// ---------------------------------------------------------------------------
__global__ __launch_bounds__(64)
void build_xcat_kernel(const float* __restrict__ ent_emb,
                       const float* __restrict__ rel_emb,
                       const int* __restrict__ cand_rel_t,
                       const int* __restrict__ cand_ent_t,
                       const int* __restrict__ parent,
                       const int* __restrict__ choice,
                       const float* __restrict__ h_prev,
                       const float* __restrict__ c_prev,
                       float* __restrict__ xcat,
                       float* __restrict__ c_par)
{
    const int m   = blockIdx.x;                // 0..1023
    const int bq  = m >> 5;
    const int p   = parent[m];
    const int ch  = choice[m];
    const int src = bq * Wb + p;
    const int ri  = cand_rel_t[src * Cc + ch];
    const int ei  = cand_ent_t[src * Cc + ch];

    const float4* __restrict__ rp4 = (const float4*)(rel_emb + (size_t)ri * E);
    const float4* __restrict__ ep4 = (const float4*)(ent_emb + (size_t)ei * E);
    const float4* __restrict__ hp4 = (const float4*)(h_prev + (size_t)src * E);
    const float4* __restrict__ cp4 = (const float4*)(c_prev + (size_t)src * E);
    float4* __restrict__ x4 = (float4*)(xcat + (size_t)m * 768);
    float4* __restrict__ c4 = (float4*)(c_par + (size_t)m * E);

    const int t = threadIdx.x;                 // 0..63
    x4[t]        = rp4[t];
    x4[64 + t]   = ep4[t];
    x4[128 + t]  = hp4[t];
    c4[t]        = cp4[t];
}

// ---------------------------------------------------------------------------
// LSTM gates: g = [i | f | g | o] (1024 cols), keras-default activations.
// ---------------------------------------------------------------------------
__device__ __forceinline__ float sigm(float x) {
    return 1.f / (1.f + __expf(-x));
}

__global__ __launch_bounds__(256)
void lstm_ew_kernel(const float* __restrict__ g,
                    const float* __restrict__ c_par,
                    float* __restrict__ h_out,
                    float* __restrict__ c_out)
{
    const int m = blockIdx.x;
    const int e = threadIdx.x;                 // 0..255
    const float* __restrict__ gr = g + (size_t)m * 1024;
    const float i_  = sigm(gr[e]);
    const float f_  = sigm(gr[256 + e]);
    const float gg  = tanhf(gr[512 + e]);
    const float o_  = sigm(gr[768 + e]);
    const float c2  = f_ * c_par[(size_t)m * E + e] + i_ * gg;
    c_out[(size_t)m * E + e] = c2;
    h_out[(size_t)m * E + e] = o_ * tanhf(c2);
}

// ---------------------------------------------------------------------------
// Host orchestration: 8 sequential steps, all on `stream` (graph-capturable).
// ---------------------------------------------------------------------------
extern "C" void kernel_launch(void* const* d_in, const int* in_sizes, int n_in,
                              void* d_out, int out_size, void* d_ws, size_t ws_size,
                              hipStream_t stream)
{
    const float* ent_emb  = (const float*)d_in[0];   // [500000, 256]
    const float* rel_emb  = (const float*)d_in[1];   // [1024, 256]
    const int*   cand_rel = (const int*)  d_in[2];   // [T,B,W,C]
    const int*   cand_ent = (const int*)  d_in[3];   // [T,B,W,C]
    const int*   start    = (const int*)  d_in[4];   // [B]
    const float* Wi       = (const float*)d_in[5];   // [512,1024]
    const float* Wh       = (const float*)d_in[6];   // [256,1024]
    const float* b_lstm   = (const float*)d_in[7];   // [1024]
    const float* W1       = (const float*)d_in[8];   // [512,512]
    const float* b1       = (const float*)d_in[9];   // [512]
    const float* W2       = (const float*)d_in[10];  // [512,512]
    const float* b2       = (const float*)d_in[11];  // [512]
    float* out = (float*)d_out;                      // hT [32,32,256]

    // ---- workspace layout (~22.3 MB of fp32 + a few KB of ints) ----
    float* ws = (float*)d_ws;
    size_t off = 0;
    auto alloc = [&](size_t n) { float* p = ws + off; off += n; return p; };
    float* h_a   = alloc((size_t)Mrows * E);
    float* h_b   = alloc((size_t)Mrows * E);
    float* c_a   = alloc((size_t)Mrows * E);
    float* c_b   = alloc((size_t)Mrows * E);
    float* c_par = alloc((size_t)Mrows * E);
    float* fin   = alloc((size_t)Mrows * 512);
    float* f1    = alloc((size_t)Mrows * 512);
    float* feat  = alloc((size_t)Mrows * 512);
    float* probs = alloc((size_t)Mrows * Cc);
    float* xcat  = alloc((size_t)Mrows * 768);
    float* gbuf  = alloc((size_t)Mrows * 1024);
    float* Wcat  = alloc((size_t)768 * 1024);
    int* cur_a   = (int*)alloc(Mrows);
    int* cur_b   = (int*)alloc(Mrows);
    int* parent  = (int*)alloc(Mrows);
    int* choice  = (int*)alloc(Mrows);
    (void)ws_size; (void)in_sizes; (void)n_in; (void)out_size;

    // one-time per launch: pack [Wi;Wh], init h/c/cur
    pack_w_kernel<<<(768 * 1024 + 255) / 256, 256, 0, stream>>>(Wi, Wh, Wcat);
    init_kernel<<<(Mrows * E + 255) / 256, 256, 0, stream>>>(start, h_a, c_a, cur_a);

    for (int t = 0; t < T_; ++t) {
        const int* crt = cand_rel + (size_t)t * Bq * Wb * Cc;
        const int* cet = cand_ent + (size_t)t * Bq * Wb * Cc;
        float* h_prev = (t & 1) ? h_b : h_a;
        float* c_prev = (t & 1) ? c_b : c_a;
        int*   curp   = (t & 1) ? cur_b : cur_a;
        float* h_next = (t == T_ - 1) ? out : ((t & 1) ? h_a : h_b);
        float* c_next = (t & 1) ? c_a : c_b;
        int*   curn   = (t & 1) ? cur_a : cur_b;

        // prior MLP: feat = relu(relu([ent_emb[cur], h] @ W1 + b1) @ W2 + b2)
        build_fin_kernel<<<Mrows, 64, 0, stream>>>(ent_emb, curp, h_prev, fin);
        gemm_bias_act_kernel<<<(Mrows / TM) * (512 / TN), 256, 0, stream>>>(
            fin, W1, b1, f1, Mrows, 512, 512, 1);
        gemm_bias_act_kernel<<<(Mrows / TM) * (512 / TN), 256, 0, stream>>>(
            f1, W2, b2, feat, Mrows, 512, 512, 1);

        // candidate scores + softmax (the HBM-gather-bound phase)
        scores_softmax_kernel<<<Mrows, 256, 0, stream>>>(
            ent_emb, rel_emb, crt, cet, feat, probs);

        // per-query top-32 beam selection
        topk_kernel<<<Bq, 256, 0, stream>>>(probs, cet, parent, choice, curn);

        // LSTM: g = [x_in, h_parent] @ [Wi;Wh] + b, then gates
        build_xcat_kernel<<<Mrows, 64, 0, stream>>>(
            ent_emb, rel_emb, crt, cet, parent, choice, h_prev, c_prev,
            xcat, c_par);
        gemm_bias_act_kernel<<<(Mrows / TM) * (1024 / TN), 256, 0, stream>>>(
            xcat, Wcat, b_lstm, gbuf, Mrows, 1024, 768, 0);
        lstm_ew_kernel<<<Mrows, 256, 0, stream>>>(gbuf, c_par, h_next, c_next);
    }
}